// SphericalWindowTokenizer_90374701842603
// MI455X (gfx1250) — compile-verified
//
#include <hip/hip_runtime.h>
#include <hip/hip_bf16.h>
#include <math.h>

#define THETA_BINS 24
#define PHI_BINS 48
#define RADIAL_BINS 4
#define NBINS (THETA_BINS * PHI_BINS * RADIAL_BINS)   /* 4608 */
#define BIN_STRIDE 5                                   /* f0..f3, count */
#define HIST_FLOATS (NBINS * BIN_STRIDE)               /* 23040 */
#define E_DIM 64
#define PI_F 3.14159265358979323846f

typedef __attribute__((ext_vector_type(2))) float v2f;
typedef __attribute__((ext_vector_type(8))) float v8f;

// ---------------------------------------------------------------------------
// Pass 0: zero workspace (rmax bits + global bins). Must run every call.
// ---------------------------------------------------------------------------
__global__ void swt_zero_ws(unsigned* __restrict__ ws, int n) {
    int i = blockIdx.x * blockDim.x + threadIdx.x;
    if (i < n) ws[i] = 0u;
}

// ---------------------------------------------------------------------------
// Pass 1: global max of r = sqrt(x^2+y^2+z^2+1e-9). r >= 0 so float bit
// pattern is monotonic under unsigned compare -> atomicMax on bits.
// 4 points per thread via 3x float4 (global_load_b128) loads.
// ---------------------------------------------------------------------------
__global__ void swt_rmax(const float* __restrict__ pts,
                         unsigned* __restrict__ rmax_bits, int npts) {
    __shared__ unsigned smax;
    if (threadIdx.x == 0) smax = 0u;
    __syncthreads();

    float lmax = 0.0f;
    int nquad = npts >> 2;
    int stride = gridDim.x * blockDim.x;
    for (int q = blockIdx.x * blockDim.x + threadIdx.x; q < nquad; q += stride) {
        const float4* p4 = (const float4*)(pts + (size_t)q * 12);
        float4 p0 = p4[0], p1 = p4[1], p2 = p4[2];
        float xs[4] = {p0.x, p0.w, p1.z, p2.y};
        float ys[4] = {p0.y, p1.x, p1.w, p2.z};
        float zs[4] = {p0.z, p1.y, p2.x, p2.w};
        #pragma unroll
        for (int j = 0; j < 4; ++j) {
            float r = sqrtf(xs[j]*xs[j] + ys[j]*ys[j] + zs[j]*zs[j] + 1e-9f);
            lmax = fmaxf(lmax, r);
        }
    }
    // tail points (npts not multiple of 4)
    if (blockIdx.x == 0 && threadIdx.x < (npts & 3)) {
        int i = (nquad << 2) + threadIdx.x;
        float x = pts[i*3+0], y = pts[i*3+1], z = pts[i*3+2];
        lmax = fmaxf(lmax, sqrtf(x*x + y*y + z*z + 1e-9f));
    }

    atomicMax(&smax, __float_as_uint(lmax));
    __syncthreads();
    if (threadIdx.x == 0) atomicMax(rmax_bits, smax);
}

// ---------------------------------------------------------------------------
// Pass 2: binning + per-block LDS histogram (92160 B of the 320 KB WGP LDS),
// flush once per block with global float atomics. Writes window_id as int4.
// ---------------------------------------------------------------------------
__global__ void swt_hist(const float* __restrict__ pts,
                         const float* __restrict__ feats,
                         const unsigned* __restrict__ rmax_bits,
                         float* __restrict__ gbins,
                         int* __restrict__ wid_out, int npts) {
    __shared__ float sb[HIST_FLOATS];
    for (int i = threadIdx.x; i < HIST_FLOATS; i += blockDim.x) sb[i] = 0.0f;
    __syncthreads();

    float rmax = __uint_as_float(*rmax_bits);
    float rinv = 1.0f / (rmax + 1e-6f);
    const float ang_scale = (float)THETA_BINS / PI_F;   // also == PHI_BINS/(2*pi)

    int nquad = npts >> 2;
    int stride = gridDim.x * blockDim.x;
    for (int q = blockIdx.x * blockDim.x + threadIdx.x; q < nquad; q += stride) {
        int qn = q + stride;
        if (qn < nquad) {
            __builtin_prefetch(pts + (size_t)qn * 12, 0, 0);
            __builtin_prefetch(feats + (size_t)qn * 16, 0, 0);
        }
        const float4* p4 = (const float4*)(pts + (size_t)q * 12);
        float4 p0 = p4[0], p1 = p4[1], p2 = p4[2];
        float xs[4] = {p0.x, p0.w, p1.z, p2.y};
        float ys[4] = {p0.y, p1.x, p1.w, p2.z};
        float zs[4] = {p0.z, p1.y, p2.x, p2.w};

        int wid[4];
        #pragma unroll
        for (int j = 0; j < 4; ++j) {
            float x = xs[j], y = ys[j], z = zs[j];
            float r = sqrtf(x*x + y*y + z*z + 1e-9f);
            float ct = fminf(fmaxf(z / r, -1.0f), 1.0f);
            float theta = acosf(ct);
            float phi = atan2f(y, x);
            int tb = (int)(theta * ang_scale);
            tb = tb < 0 ? 0 : (tb > THETA_BINS-1 ? THETA_BINS-1 : tb);
            int pb = (int)((phi + PI_F) * ang_scale);
            pb = pb < 0 ? 0 : (pb > PHI_BINS-1 ? PHI_BINS-1 : pb);
            float rn = fminf(fmaxf(r * rinv, 0.0f), 1.0f);
            int rb = (int)(rn * (float)RADIAL_BINS);
            rb = rb < 0 ? 0 : (rb > RADIAL_BINS-1 ? RADIAL_BINS-1 : rb);
            wid[j] = tb * (PHI_BINS*RADIAL_BINS) + pb * RADIAL_BINS + rb;
        }

        const float4* f4 = (const float4*)feats + (size_t)q * 4;
        #pragma unroll
        for (int j = 0; j < 4; ++j) {
            float4 f = f4[j];
            float* base = sb + wid[j] * BIN_STRIDE;
            atomicAdd(base + 0, f.x);
            atomicAdd(base + 1, f.y);
            atomicAdd(base + 2, f.z);
            atomicAdd(base + 3, f.w);
            atomicAdd(base + 4, 1.0f);
        }
        ((int4*)wid_out)[q] = make_int4(wid[0], wid[1], wid[2], wid[3]);
    }
    // tail points
    if (blockIdx.x == 0 && threadIdx.x < (npts & 3)) {
        int i = (nquad << 2) + threadIdx.x;
        float x = pts[i*3+0], y = pts[i*3+1], z = pts[i*3+2];
        float r = sqrtf(x*x + y*y + z*z + 1e-9f);
        float ct = fminf(fmaxf(z / r, -1.0f), 1.0f);
        int tb = (int)(acosf(ct) * ang_scale);
        tb = tb < 0 ? 0 : (tb > THETA_BINS-1 ? THETA_BINS-1 : tb);
        int pb = (int)((atan2f(y, x) + PI_F) * ang_scale);
        pb = pb < 0 ? 0 : (pb > PHI_BINS-1 ? PHI_BINS-1 : pb);
        float rn = fminf(fmaxf(r * rinv, 0.0f), 1.0f);
        int rb = (int)(rn * (float)RADIAL_BINS);
        rb = rb < 0 ? 0 : (rb > RADIAL_BINS-1 ? RADIAL_BINS-1 : rb);
        int w = tb * (PHI_BINS*RADIAL_BINS) + pb * RADIAL_BINS + rb;
        float* base = sb + w * BIN_STRIDE;
        atomicAdd(base + 0, feats[i*4+0]);
        atomicAdd(base + 1, feats[i*4+1]);
        atomicAdd(base + 2, feats[i*4+2]);
        atomicAdd(base + 3, feats[i*4+3]);
        atomicAdd(base + 4, 1.0f);
        wid_out[i] = w;
    }

    __syncthreads();
    for (int i = threadIdx.x; i < HIST_FLOATS; i += blockDim.x) {
        float v = sb[i];
        if (v != 0.0f) atomicAdd(gbins + i, v);
    }
}

// ---------------------------------------------------------------------------
// Pass 3: embedded = (bins/cnt) @ W + b with V_WMMA_F32_16X16X4_F32.
// K = 4 = feature dim, so one WMMA per 16x16 output tile.
// One wave (32 lanes) per tile; 288 M-tiles x 4 N-tiles = 1152 waves.
//   A (16x4 f32):  lane<16 holds {K0,K1} of row M=lane; lane>=16 holds {K2,K3}.
//   C/D (16x16):   VGPR i -> M=i (lanes 0-15) / M=i+8 (lanes 16-31), N=lane%16.
// ---------------------------------------------------------------------------
__global__ void swt_embed(const float* __restrict__ gbins,
                          const float* __restrict__ W,
                          const float* __restrict__ bias,
                          float* __restrict__ out) {
    int tile  = blockIdx.x;
    int tileM = tile >> 2;          // 0..287
    int tileN = tile & 3;           // 0..3
    int lane  = threadIdx.x;        // wave32
    int half  = lane >> 4;          // 0: K=0,1   1: K=2,3
    int l16   = lane & 15;
    int k0    = half * 2;

    // A fragment: normalized tokens
    int m = tileM * 16 + l16;
    const float* row = gbins + m * BIN_STRIDE;
    float inv = 1.0f / fmaxf(row[4], 1.0f);
    v2f a;
    a.x = row[k0 + 0] * inv;
    a.y = row[k0 + 1] * inv;

    // B fragment: W is (4 x 64) row-major
    int n = tileN * 16 + l16;
    v2f bf;
    bf.x = W[(k0 + 0) * E_DIM + n];
    bf.y = W[(k0 + 1) * E_DIM + n];

    v8f c = {};
    c = __builtin_amdgcn_wmma_f32_16x16x4_f32(
            /*neg_a=*/false, a, /*neg_b=*/false, bf,
            /*c_mod=*/(short)0, c, /*reuse_a=*/false, /*reuse_b=*/false);

    float bv = bias[n];
    #pragma unroll
    for (int i = 0; i < 8; ++i) {
        int mrow = tileM * 16 + i + half * 8;
        out[mrow * E_DIM + n] = c[i] + bv;
    }
}

// ---------------------------------------------------------------------------
extern "C" void kernel_launch(void* const* d_in, const int* in_sizes, int n_in,
                              void* d_out, int out_size, void* d_ws, size_t ws_size,
                              hipStream_t stream) {
    const float* pts   = (const float*)d_in[0];   // (N,3)
    const float* feats = (const float*)d_in[1];   // (N,4)
    const float* W     = (const float*)d_in[2];   // (4,64)
    const float* bias  = (const float*)d_in[3];   // (64,)
    int npts = in_sizes[0] / 3;

    float* embedded = (float*)d_out;                       // 4608*64 floats
    int*   wid_out  = (int*)(embedded + NBINS * E_DIM);    // then N ints

    unsigned* rmax_bits = (unsigned*)d_ws;                 // ws[0]
    float*    gbins     = (float*)d_ws + 16;               // ws[16..16+23040)

    int zeroN = 16 + HIST_FLOATS;
    swt_zero_ws<<<(zeroN + 255) / 256, 256, 0, stream>>>((unsigned*)d_ws, zeroN);

    swt_rmax<<<2048, 256, 0, stream>>>(pts, rmax_bits, npts);

    swt_hist<<<512, 256, 0, stream>>>(pts, feats, rmax_bits, gbins, wid_out, npts);

    swt_embed<<<(NBINS / 16) * (E_DIM / 16), 32, 0, stream>>>(gbins, W, bias, embedded);
}